// KANDecoderHead_10599979287094
// MI455X (gfx1250) — compile-verified
//
#include <hip/hip_runtime.h>

typedef float v2f __attribute__((ext_vector_type(2)));
typedef float v8f __attribute__((ext_vector_type(8)));
typedef int   v4i __attribute__((vector_size(16)));              // matches builtin param
typedef __attribute__((address_space(1))) v4i* v4i_gptr;          // global
typedef __attribute__((address_space(3))) v4i* v4i_lptr;          // LDS

#define IN_CH    256
#define HIDDEN   128
#define NCLS     19
#define NPIX     2048      // B*H*W = 2*32*32
#define PIX_PER_B 1024
#define GROUPS   32
#define CPG      8         // channels per group

// ---------------------------------------------------------------------------
// Kernel 1: 1x1 conv as GEMM  feat[n, o] = sum_c x[b, c, p] * conv_w[o, c] + conv_b[o]
// One wave computes a 16x16 tile via V_WMMA_F32_16X16X4_F32.
// A(M=pixel-row, K=c): lanes 0-15 M=lane, VGPR0/1 = K=(hi*2), K=(hi*2+1)
// B(K=c, N=o)        : lanes 0-15 N=lane, VGPR0/1 = K=(hi*2), K=(hi*2+1)
// C/D: VGPR r, lanes 0-15 -> (M=r, N=lane); lanes 16-31 -> (M=r+8, N=lane-16)
// ---------------------------------------------------------------------------
__global__ void conv_gemm(const float* __restrict__ x,
                          const float* __restrict__ w,
                          const float* __restrict__ bias,
                          float* __restrict__ feat) {
  const int lane  = threadIdx.x & 31;
  const int wave  = threadIdx.x >> 5;
  const int n0    = blockIdx.x * 16;                 // pixel-row tile
  const int o0    = (blockIdx.y * 8 + wave) * 16;    // out-channel tile
  const int b     = n0 / PIX_PER_B;
  const int p0    = n0 % PIX_PER_B;
  const int ln    = lane & 15;
  const int kh    = (lane >> 4) * 2;
  const int mbase = (lane >> 4) * 8;

  v8f acc;
  const float bv = bias[o0 + ln];
#pragma unroll
  for (int r = 0; r < 8; ++r) acc[r] = bv;

  // element (pixel n0+ln, channel c) lives at xa + c*1024
  const float* xa = x + (size_t)b * IN_CH * PIX_PER_B + (p0 + ln);
  const float* wb = w + (size_t)(o0 + ln) * IN_CH;   // conv_w row for this lane's o

  for (int k0 = 0; k0 < IN_CH; k0 += 4) {
    v2f a, bm;
    a.x = xa[(size_t)(k0 + kh + 0) * PIX_PER_B];
    a.y = xa[(size_t)(k0 + kh + 1) * PIX_PER_B];
    bm  = *(const v2f*)(wb + k0 + kh);
    acc = __builtin_amdgcn_wmma_f32_16x16x4_f32(false, a, false, bm,
                                                (short)0, acc, false, false);
  }

#pragma unroll
  for (int r = 0; r < 8; ++r) {
    const int n = n0 + mbase + r;
    feat[(size_t)n * IN_CH + (o0 + ln)] = acc[r];
  }
}

// ---------------------------------------------------------------------------
// Kernel 2: GroupNorm over (C/G channels x all pixels of one batch image).
// One block per (b, g). Population variance (ddof=0) like jnp.var.
// ---------------------------------------------------------------------------
__global__ void group_norm(float* __restrict__ feat,
                           const float* __restrict__ gw,
                           const float* __restrict__ gb) {
  const int b = blockIdx.x >> 5;
  const int g = blockIdx.x & 31;
  const int t = threadIdx.x;
  __shared__ float s_sum[256];
  __shared__ float s_sq[256];

  float* base = feat + (size_t)b * PIX_PER_B * IN_CH + g * CPG;
  float sum = 0.f, sq = 0.f;
  for (int e = t; e < CPG * PIX_PER_B; e += 256) {
    const int p = e >> 3, j = e & 7;
    const float v = base[(size_t)p * IN_CH + j];
    sum += v; sq += v * v;
  }
  s_sum[t] = sum; s_sq[t] = sq;
  __syncthreads();
  for (int s = 128; s > 0; s >>= 1) {
    if (t < s) { s_sum[t] += s_sum[t + s]; s_sq[t] += s_sq[t + s]; }
    __syncthreads();
  }
  const float inv_n = 1.0f / (float)(CPG * PIX_PER_B);
  const float mean  = s_sum[0] * inv_n;
  const float var   = s_sq[0] * inv_n - mean * mean;
  const float rstd  = rsqrtf(var + 1e-5f);

  for (int e = t; e < CPG * PIX_PER_B; e += 256) {
    const int p = e >> 3, j = e & 7;
    const int c = g * CPG + j;
    float* p_ = base + (size_t)p * IN_CH + j;
    *p_ = (*p_ - mean) * rstd * gw[c] + gb[c];
  }
}

// ---------------------------------------------------------------------------
// Kernels 3/4: fused Rational-KAN layer.
//   out[n,o] = sum_c f[n,c]*W[o,c] + bias[o]
//            + sum_c P(f[n,c]; Pc[o,c,:]) / (Q(f[n,c]; Qc[o,c,:]) + 1e-6)
// Tile of 16 activation rows is staged into LDS with the CDNA5 async
// global->LDS path (ASYNCcnt, no VGPR round-trip); base matmul runs
// V_WMMA_F32_16X16X4_F32 on it, and the rational sum accumulates directly
// into the C-fragment (lane = column o, VGPR r = row).
// ---------------------------------------------------------------------------
template <int I, int O, int WAVES, bool FINAL>
__global__ void kan_layer(const float* __restrict__ fin,   // (NPIX, I)
                          const float* __restrict__ P,     // (O, I, 4)
                          const float* __restrict__ Q,     // (O, I, 3)
                          const float* __restrict__ W,     // (O, I)
                          const float* __restrict__ bias,  // (O)
                          float* __restrict__ out) {
  const int lane  = threadIdx.x & 31;
  const int wave  = threadIdx.x >> 5;
  const int n0    = blockIdx.x * 16;
  const int o0    = wave * 16;
  const int ln    = lane & 15;
  const int hi    = lane >> 4;
  const int kh    = hi * 2;
  const int mbase = hi * 8;

  // +4 pad: rows stay 16B-aligned for b128 async writes; stride mod 64 = 4
  // skews banks so the 16 A-fragment rows hit 16 distinct banks.
  __shared__ float ft[16][I + 4];

#if __has_builtin(__builtin_amdgcn_global_load_async_to_lds_b128)
  // Async global->LDS (GLOBAL_LOAD_ASYNC_TO_LDS_B128), tracked by ASYNCcnt.
  // AS casts go through integers: AS1 VA == generic VA for global memory;
  // AS3 pointer is the low 32 bits of the generic shared address (HW LDS
  // aperture truncation rule).
  for (int e = threadIdx.x; e < 16 * (I / 4); e += WAVES * 32) {
    const int m = e / (I / 4);
    const int c = (e % (I / 4)) * 4;
    v4i_gptr gsrc = (v4i_gptr)(unsigned long long)(fin + (size_t)(n0 + m) * I + c);
    v4i_lptr ldst = (v4i_lptr)(unsigned int)(unsigned long long)&ft[m][c];
    __builtin_amdgcn_global_load_async_to_lds_b128(gsrc, ldst,
                                                   /*offset=*/0, /*cpol=*/0);
  }
#if __has_builtin(__builtin_amdgcn_s_wait_asynccnt)
  __builtin_amdgcn_s_wait_asynccnt(0);
#else
  asm volatile("s_wait_asynccnt 0x0" ::: "memory");
#endif
  __syncthreads();
#else
  for (int e = threadIdx.x; e < 16 * I; e += WAVES * 32) {
    const int m = e / I, c = e % I;
    ft[m][c] = fin[(size_t)(n0 + m) * I + c];
  }
  __syncthreads();
#endif

  const int o  = o0 + ln;
  const int oc = (o < O) ? o : (O - 1);   // clamped index for safe loads

  v8f acc;
  const float bv = bias[oc];
#pragma unroll
  for (int r = 0; r < 8; ++r) acc[r] = bv;

  // --- base matmul: 16x16 tile, K stepped by 4 ---
  const float* wrow = W + (size_t)oc * I;
  for (int k0 = 0; k0 < I; k0 += 4) {
    v2f a  = *(const v2f*)&ft[ln][k0 + kh];
    v2f bm = *(const v2f*)(wrow + k0 + kh);
    acc = __builtin_amdgcn_wmma_f32_16x16x4_f32(false, a, false, bm,
                                                (short)0, acc, false, false);
  }

  // --- rational part: each lane owns column o, rows mbase..mbase+7 ---
  const float* Pr = P + (size_t)oc * I * 4;
  const float* Qr = Q + (size_t)oc * I * 3;
  for (int c = 0; c < I; ++c) {
    __builtin_prefetch(Pr + (c + 8) * 4, 0, 0);      // global_prefetch_b8
    const float4 pc = *(const float4*)(Pr + c * 4);  // b128 coeff load
    const float q1c = Qr[c * 3 + 0];
    const float q2c = Qr[c * 3 + 1];
    const float q3c = Qr[c * 3 + 2];
#pragma unroll
    for (int r = 0; r < 8; ++r) {
      const float xv   = ft[mbase + r][c];           // broadcast within half-wave
      const float pnum = fmaf(fmaf(fmaf(pc.w, xv, pc.z), xv, pc.y), xv, pc.x);
      const float qden = fmaf(fmaf(fmaf(q3c, xv, q2c), xv, q1c), xv, 1.0f);
      acc[r] += pnum / (qden + 1e-6f);
    }
  }

  // --- store ---
  if (o < O) {
    if (!FINAL) {
#pragma unroll
      for (int r = 0; r < 8; ++r)
        out[(size_t)(n0 + mbase + r) * O + o] = acc[r];
    } else {
#pragma unroll
      for (int r = 0; r < 8; ++r) {
        const int n  = n0 + mbase + r;
        const int bb = n / PIX_PER_B;
        const int pp = n % PIX_PER_B;
        // (B, NCLS, H, W) layout
        out[((size_t)bb * O + o) * PIX_PER_B + pp] = acc[r];
      }
    }
  }
}

// ---------------------------------------------------------------------------
extern "C" void kernel_launch(void* const* d_in, const int* in_sizes, int n_in,
                              void* d_out, int out_size, void* d_ws, size_t ws_size,
                              hipStream_t stream) {
  const float* x      = (const float*)d_in[0];
  const float* conv_w = (const float*)d_in[1];
  const float* conv_b = (const float*)d_in[2];
  const float* gn_w   = (const float*)d_in[3];
  const float* gn_b   = (const float*)d_in[4];
  const float* P1     = (const float*)d_in[5];
  const float* Q1     = (const float*)d_in[6];
  const float* bw1    = (const float*)d_in[7];
  const float* bb1    = (const float*)d_in[8];
  const float* P2     = (const float*)d_in[9];
  const float* Q2     = (const float*)d_in[10];
  const float* bw2    = (const float*)d_in[11];
  const float* bb2    = (const float*)d_in[12];
  float* out  = (float*)d_out;

  float* feat = (float*)d_ws;                        // NPIX * IN_CH  = 2 MB
  float* h    = feat + (size_t)NPIX * IN_CH;         // NPIX * HIDDEN = 1 MB

  // 1) 1x1 conv GEMM: 128 row-tiles x 16 col-tiles (8 waves/block x 2 blocks.y)
  conv_gemm<<<dim3(NPIX / 16, 2), 256, 0, stream>>>(x, conv_w, conv_b, feat);
  // 2) GroupNorm, one block per (batch, group)
  group_norm<<<2 * GROUPS, 256, 0, stream>>>(feat, gn_w, gn_b);
  // 3) kan1: 2048x256 -> 2048x128, 8 o-tiles per block
  kan_layer<IN_CH, HIDDEN, 8, false><<<NPIX / 16, 8 * 32, 0, stream>>>(
      feat, P1, Q1, bw1, bb1, h);
  // 4) kan2: 2048x128 -> 2048x19 (2 o-tiles, masked stores), writes (B,19,H,W)
  kan_layer<HIDDEN, NCLS, 2, true><<<NPIX / 16, 2 * 32, 0, stream>>>(
      h, P2, Q2, bw2, bb2, out);
}